// distance2edge_selfLoop_7541962572511
// MI455X (gfx1250) — compile-verified
//
#include <hip/hip_runtime.h>
#include <stdint.h>

// distance2edge with self-loops: edges = argwhere(dist < 0.01), row-major,
// padded with -1 to MAX_EDGES. Output = rows[MAXE] ++ cols[MAXE] (int32).

#define N_DIM   12288
#define NN      (N_DIM * N_DIM)          // 150,994,944
#define THR     0.01f

#define CHUNK   2048                     // floats per TDM tile (8 KB)
#define WPB     4                        // waves per block (128 threads)
#define CPW     8                        // chunks per wave
#define NCHUNK  (NN / CHUNK)             // 73,728
#define P1_BLOCKS (NCHUNK / (WPB * CPW)) // 2,304
#define SCAN_T  1024
#define SCAN_PER (NCHUNK / SCAN_T)       // 72

static_assert(NN % CHUNK == 0, "chunking");
static_assert(NCHUNK % (WPB * CPW) == 0, "grid");
static_assert(NCHUNK % SCAN_T == 0, "scan");

typedef unsigned int u32;
typedef unsigned long long u64;
typedef u32 u32x4_t __attribute__((ext_vector_type(4)));
typedef u32 u32x8_t __attribute__((ext_vector_type(8)));

// ---------------------------------------------------------------------------
// TDM: issue tensor_load_to_lds for one 1-D tile of CHUNK f32 (global -> LDS).
// D# per CDNA5 ISA ch.8: group0 = {flags/count, lds_addr, global_addr, type=2},
// group1 = {data_size=4B, tensor_dim0=tile_dim0=CHUNK, dims/strides}.
// 2-operand form => VADDR2/VADDR3 = NULL (<=2D tensor). Tracked by TENSORcnt.
// ---------------------------------------------------------------------------
__device__ __forceinline__ void tdm_load_tile(const float* gsrc, const float* lds_dst) {
    u64 ga  = (u64)(uintptr_t)gsrc;
    u32 lds = (u32)(uintptr_t)lds_dst;   // low 32 bits of generic ptr = LDS byte offset
    u32x4_t g0 = {
        1u,                                                  // count=1 (valid), user mode
        lds,                                                 // lds_addr [63:32]
        (u32)ga,                                             // global_addr lo
        ((u32)(ga >> 32) & 0x01FFFFFFu) | 0x80000000u        // global_addr hi | type=2
    };
    u32x8_t g1 = {
        2u << 16,                       // workgroup_mask=0 (not in cluster), data_size=2 (4B)
        ((u32)CHUNK & 0xFFFFu) << 16,   // tensor_dim0[15:0]
        ((u32)CHUNK >> 16) | (1u << 16),// tensor_dim0[31:16] | tensor_dim1=1
        ((u32)CHUNK) << 16,             // tensor_dim1 hi=0 | tile_dim0=CHUNK
        0u,                             // tile_dim1=0, tile_dim2=0 (unused)
        (u32)CHUNK,                     // tensor_dim0_stride lo
        0u, 0u                          // stride hi / tensor_dim1_stride = 0
    };
    asm volatile("tensor_load_to_lds %0, %1" :: "s"(g0), "s"(g1) : "memory");
}

// Build 64-bit predicate mask for 64 contiguous floats (16 x float4).
__device__ __forceinline__ u64 mask64_from(const float4* p) {
    u64 m = 0;
#pragma unroll
    for (int q = 0; q < 16; ++q) {
        float4 v = p[q];
        u64 nib = (u64)((v.x < THR ? 1u : 0u) | (v.y < THR ? 2u : 0u) |
                        (v.z < THR ? 4u : 0u) | (v.w < THR ? 8u : 0u));
        m |= nib << (4 * q);
    }
    return m;
}

// ---------------------------------------------------------------------------
// Pass 1: stream matrix once via double-buffered TDM tiles; emit per-chunk
// count (and 64-bit-per-lane bitmask when scratch permits).
// ---------------------------------------------------------------------------
__global__ void __launch_bounds__(WPB * 32)
pass1_count(const float* __restrict__ mat, u32* __restrict__ counts,
            u64* __restrict__ maskBuf, int use_mask) {
    __shared__ float smem[WPB][2][CHUNK];   // 64 KB static LDS
    const int lane = threadIdx.x & 31;
    const int w    = threadIdx.x >> 5;
    const int wave = blockIdx.x * WPB + w;
    const int c0   = wave * CPW;

    tdm_load_tile(mat + (size_t)c0 * CHUNK, &smem[w][0][0]);

    for (int k = 0; k < CPW; ++k) {
        const int c = c0 + k;
        if (k + 1 < CPW) {
            tdm_load_tile(mat + (size_t)(c + 1) * CHUNK, &smem[w][(k + 1) & 1][0]);
            asm volatile("s_wait_tensorcnt 0x1" ::: "memory");   // tile k landed
        } else {
            asm volatile("s_wait_tensorcnt 0x0" ::: "memory");   // last tile landed
        }
        const float* lbuf = &smem[w][k & 1][0];
        u64 m = mask64_from(reinterpret_cast<const float4*>(lbuf + lane * 64));

        u32 cnt = (u32)__popcll(m);
#pragma unroll
        for (int o = 16; o > 0; o >>= 1) cnt += __shfl_xor(cnt, o, 32);
        if (lane == 0) counts[c] = cnt;
        if (use_mask)  maskBuf[(size_t)c * 32 + lane] = m;
    }
}

// ---------------------------------------------------------------------------
// Single-block exclusive scan over NCHUNK chunk counts (73,728 u32, 288 KB).
// ---------------------------------------------------------------------------
__global__ void __launch_bounds__(SCAN_T)
scan_kernel(const u32* __restrict__ counts, u32* __restrict__ offsets) {
    __shared__ u32 sums[SCAN_T];
    const int t = threadIdx.x;
    const int base = t * SCAN_PER;
    u32 s = 0;
    for (int i = 0; i < SCAN_PER; ++i) s += counts[base + i];
    sums[t] = s;
    __syncthreads();
    for (int o = 1; o < SCAN_T; o <<= 1) {        // Hillis-Steele inclusive
        u32 v = (t >= o) ? sums[t - o] : 0u;
        __syncthreads();
        sums[t] += v;
        __syncthreads();
    }
    u32 run = sums[t] - s;                         // exclusive prefix
    for (int i = 0; i < SCAN_PER; ++i) { offsets[base + i] = run; run += counts[base + i]; }
}

// ---------------------------------------------------------------------------
// Fill entire output with -1 (argwhere padding).
// ---------------------------------------------------------------------------
__global__ void fill_kernel(int4* __restrict__ out, int n4) {
    int i = blockIdx.x * blockDim.x + threadIdx.x;
    if (i < n4) out[i] = make_int4(-1, -1, -1, -1);
}

// ---------------------------------------------------------------------------
// Pass 2: ordered emit. Predicate from bitmask (19 MB) or recomputed from the
// matrix (fallback). Wave-level shfl_up prefix orders lanes; scan result
// orders chunks; LSB-first bit walk orders elements within a lane.
// ---------------------------------------------------------------------------
__global__ void __launch_bounds__(WPB * 32)
pass2_emit(const float* __restrict__ mat, const u64* __restrict__ maskBuf,
           const u32* __restrict__ offsets, int* __restrict__ out_row,
           int* __restrict__ out_col, int use_mask, u32 maxe) {
    const int lane = threadIdx.x & 31;
    const int wave = blockIdx.x * WPB + (threadIdx.x >> 5);
    const int c0   = wave * CPW;

    for (int k = 0; k < CPW; ++k) {
        const int c = c0 + k;
        u64 m;
        if (use_mask) {
            m = maskBuf[(size_t)c * 32 + lane];
        } else {
            m = mask64_from(reinterpret_cast<const float4*>(
                    mat + (size_t)c * CHUNK + (size_t)lane * 64));
        }
        u32 cnt  = (u32)__popcll(m);
        u32 incl = cnt;
#pragma unroll
        for (int o = 1; o < 32; o <<= 1) {
            u32 v = __shfl_up(incl, o, 32);
            if (lane >= o) incl += v;
        }
        u32 pos = offsets[c] + incl - cnt;
        u32 flatBase = (u32)c * (u32)CHUNK + (u32)lane * 64u;
        while (m) {
            u32 b    = (u32)__builtin_ctzll(m);
            u32 flat = flatBase + b;
            u32 i    = flat / (u32)N_DIM;
            u32 j    = flat - i * (u32)N_DIM;
            if (pos < maxe) { out_row[pos] = (int)i; out_col[pos] = (int)j; }
            ++pos;
            m &= m - 1;
        }
    }
}

// ---------------------------------------------------------------------------
extern "C" void kernel_launch(void* const* d_in, const int* in_sizes, int n_in,
                              void* d_out, int out_size, void* d_ws, size_t ws_size,
                              hipStream_t stream) {
    const float* mat = (const float*)d_in[0];

    const u32 maxe  = (u32)(out_size / 2);      // 3,200,000
    int* out_row = (int*)d_out;
    int* out_col = out_row + maxe;

    u32* counts  = (u32*)d_ws;
    u32* offsets = counts + NCHUNK;
    u64* maskBuf = (u64*)(offsets + NCHUNK);    // 8B-aligned: 2*NCHUNK*4 = 589,824
    const size_t need_mask = (size_t)NCHUNK * 2 * sizeof(u32)
                           + (size_t)NCHUNK * 32 * sizeof(u64);   // ~19.5 MB
    const int use_mask = (ws_size >= need_mask) ? 1 : 0;

    const int n4 = out_size / 4;
    fill_kernel<<<(n4 + 255) / 256, 256, 0, stream>>>((int4*)d_out, n4);
    pass1_count<<<P1_BLOCKS, WPB * 32, 0, stream>>>(mat, counts, maskBuf, use_mask);
    scan_kernel<<<1, SCAN_T, 0, stream>>>(counts, offsets);
    pass2_emit<<<P1_BLOCKS, WPB * 32, 0, stream>>>(mat, maskBuf, offsets,
                                                   out_row, out_col, use_mask, maxe);
}